// NeuralCameraModule_1726576855928
// MI455X (gfx1250) — compile-verified
//
#include <hip/hip_runtime.h>
#include <hip/hip_bf16.h>
#include <stdint.h>

#define NS 64
#define WPB 8             // waves per block (block = 256 threads)

__global__ __launch_bounds__(256, 2)
void nerf_importance_kernel(const float* __restrict__ extrinsic,
                            const float* __restrict__ intrinsic,
                            const float* __restrict__ Rm,
                            const float* __restrict__ Tm,
                            const float* __restrict__ Sm,
                            const float* __restrict__ density,
                            const float* __restrict__ uin,
                            float* __restrict__ out,
                            int nrays) {
  __shared__ float stage[WPB][128];   // per wave: [0..63]=density row, [64..127]=u row
  __shared__ float cdfl[WPB][64];     // per wave: cdf[0..62] + pad[63]=+inf

  const int wv   = threadIdx.x >> 5;
  const int lane = threadIdx.x & 31;
  // one ray per wave -> ray is wave-uniform; pin it to an SGPR so the whole
  // camera-setup phase goes down the scalar path (s_load + SALU float).
  const int ray  = __builtin_amdgcn_readfirstlane(blockIdx.x * WPB + wv);
  if (ray >= nrays) return;

  // ---- CDNA5 async staging: one GLOBAL_LOAD_ASYNC_TO_LDS_B128 per wave ----
  // lanes 0..15 pull 16B each of density[ray,:]; lanes 16..31 pull u[ray,:].
  // Branchless addressing: select only the 64-bit base (v_cndmask pair).
  {
    const unsigned long long bd = (unsigned long long)(uintptr_t)density;
    const unsigned long long bu = (unsigned long long)(uintptr_t)uin;
    const unsigned long long base = (lane < 16) ? bd : bu;
    const unsigned long long ga =
        base + (unsigned long long)ray * (NS * 4ull) + ((lane & 15) << 4);
    const unsigned int lo = (unsigned int)(uintptr_t)(&stage[wv][0]) +
                            ((lane & 15) << 4) + ((lane >> 4) << 8);
    asm volatile("global_load_async_to_lds_b128 %0, %1, off"
                 :: "v"(lo), "v"(ga) : "memory");
  }

  // ---- camera / ray setup (wave-uniform; overlaps with the async load) ----
  const int b  = ray >> 16;       // RES*RES == 65536
  const int n  = ray & 0xFFFF;
  const int py = n >> 8;          // height index (j)
  const int px = n & 0xFF;        // width index  (i)

  const float* E  = extrinsic + b * 16;
  const float* I  = intrinsic + b * 9;
  const float* Rb = Rm + b * 9;
  const float* Tb = Tm + b * 3;
  const float  Sv = Sm[b];

  const float x  = 0.5f + px * (511.0f / 255.0f);   // linspace(0.5, 511.5, 256)
  const float y  = 0.5f + py * (511.0f / 255.0f);
  const float dx = (x - I[2]) / I[0];
  const float dy = (y - I[5]) / I[4];

  float rd[3], ro[3];
#pragma unroll
  for (int r = 0; r < 3; ++r) {
    // rot[r][c] = E[c][r];  rays_d[r] = sum_c dirs[c]*rot[r][c]
    rd[r] = dx * E[r] + dy * E[4 + r] + E[8 + r];
    // trans[r] = -sum_j E[j][r]*E[j][3]
    ro[r] = -(E[r] * E[3] + E[4 + r] * E[7] + E[8 + r] * E[11]);
  }
  const float norm_rd = sqrtf(rd[0]*rd[0] + rd[1]*rd[1] + rd[2]*rd[2]);

  float oc[3], dc[3];
  const float invS = 1.0f / Sv;
#pragma unroll
  for (int i = 0; i < 3; ++i) {
    oc[i] = (Rb[i]*(ro[0]-Tb[0]) + Rb[3+i]*(ro[1]-Tb[1]) + Rb[6+i]*(ro[2]-Tb[2])) * invS;
    dc[i] = (Rb[i]*rd[0]         + Rb[3+i]*rd[1]         + Rb[6+i]*rd[2]) * invS;
  }
  const float normd = sqrtf(dc[0]*dc[0] + dc[1]*dc[1] + dc[2]*dc[2]);

  float near0 = -3.4e38f, far0 = 3.4e38f;
#pragma unroll
  for (int i = 0; i < 3; ++i) {
    float v = dc[i] / normd;
    if (v < 1e-5f  && v > -1e-10f) v =  1e-5f;
    if (v > -1e-5f && v <  1e-10f) v = -1e-5f;
    const float ta = (-1.0f - oc[i]) / v;
    const float tb = ( 1.0f - oc[i]) / v;
    near0 = fmaxf(near0, fminf(ta, tb));
    far0  = fminf(far0,  fmaxf(ta, tb));
  }
  const bool  hit   = near0 < far0;
  const float invnd = 1.0f / normd;
  const float nearv = hit ? near0 * invnd : 0.5f;
  const float farv  = hit ? far0  * invnd : 1.5f;

  const float zstep   = (farv - nearv) * (1.0f / 63.0f);   // uniform z spacing
  const float dist    = zstep * 100.0f * norm_rd;          // dists[s], s<63
  const float distLst = 1e10f * norm_rd;                   // dists[63]

  // ---- consume staged rows ----
  asm volatile("s_wait_asynccnt 0" ::: "memory");
  const float den0 = fmaxf(stage[wv][2*lane    ], 0.0f);
  const float den1 = fmaxf(stage[wv][2*lane + 1], 0.0f);

  const float a0 = 1.0f - __expf(-den0 * dist);
  const float a1 = 1.0f - __expf(-den1 * ((lane == 31) ? distLst : dist));

  // exclusive cumprod of (1 - alpha + 1e-10) via wave multiply-scan
  const float f0 = 1.0f - a0 + 1e-10f;
  const float f1 = 1.0f - a1 + 1e-10f;
  float P = f0 * f1;
#pragma unroll
  for (int d = 1; d < 32; d <<= 1) {
    float t = __shfl_up(P, d, 32);
    if (lane >= d) P *= t;
  }
  float exclP = __shfl_up(P, 1, 32);
  if (lane == 0) exclP = 1.0f;
  const float w0 = a0 * exclP;          // weights[2L]
  const float w1 = a1 * exclP * f0;     // weights[2L+1]

  // pdf over weights[1..62] (62 entries) -> cdf (63 entries) via wave add-scan
  const float wdn = __shfl_down(w0, 1, 32);                 // weights[2L+2]
  const float wi0 = (lane <= 30) ? (w1  + 1e-5f) : 0.0f;    // w index 2L
  const float wi1 = (lane <= 30) ? (wdn + 1e-5f) : 0.0f;    // w index 2L+1
  const float ps  = wi0 + wi1;
  float Sc = ps;
#pragma unroll
  for (int d = 1; d < 32; d <<= 1) {
    float t = __shfl_up(Sc, d, 32);
    if (lane >= d) Sc += t;
  }
  const float total = __shfl(Sc, 31, 32);
  const float invT  = 1.0f / total;
  const float Sxcl  = Sc - ps;

  if (lane == 0)  cdfl[wv][0] = 0.0f;
  if (lane <= 30) {
    cdfl[wv][2*lane + 1] = (Sxcl + wi0) * invT;
    cdfl[wv][2*lane + 2] = Sc * invT;
  }
  if (lane == 31) cdfl[wv][63] = 3.4e38f;   // pad: +inf sentinel for the search
  // same-wave LDS ops are in-order: writes above are visible to reads below.

  // ---- inverse-CDF sampling: branchless rank search over 64-entry LDS table ----
  const float* cp = cdfl[wv];
  float smp[2];
#pragma unroll
  for (int r = 0; r < 2; ++r) {
    const float uu = stage[wv][64 + 2*lane + r];
    int idx = 0;
#pragma unroll
    for (int h = 32; h >= 1; h >>= 1)
      if (cp[idx + h - 1] <= uu) idx += h;      // searchsorted(..., 'right')
    const int below = (idx - 1 < 0) ? 0 : idx - 1;
    const int above = (idx > 62) ? 62 : idx;
    const float cb = cp[below], ca = cp[above];
    float dnm = ca - cb;
    dnm = (dnm < 1e-5f) ? 1.0f : dnm;
    const float t  = (uu - cb) / dnm;
    const float bb = nearv + zstep * ((float)below + 0.5f);  // bins are affine in index
    const float ba = nearv + zstep * ((float)above + 0.5f);
    smp[r] = bb + t * (ba - bb);
  }

  // ---- 64-element bitonic sort across the wave (2 regs/lane) ----
  float s0 = smp[0], s1 = smp[1];
#pragma unroll
  for (int k = 2; k <= 64; k <<= 1) {
#pragma unroll
    for (int j = k >> 1; j >= 2; j >>= 1) {
      const int lm = j >> 1;
      const int e0 = 2*lane, e1 = 2*lane + 1;
      float o0 = __shfl_xor(s0, lm, 32);
      const bool kp0 = ((e0 & k) == 0) == ((e0 & j) == 0);
      s0 = kp0 ? fminf(s0, o0) : fmaxf(s0, o0);
      float o1 = __shfl_xor(s1, lm, 32);
      const bool kp1 = ((e1 & k) == 0) == ((e1 & j) == 0);
      s1 = kp1 ? fminf(s1, o1) : fmaxf(s1, o1);
    }
    const bool up = (((2*lane) & k) == 0);
    const float mn = fminf(s0, s1), mx = fmaxf(s0, s1);
    s0 = up ? mn : mx;
    s1 = up ? mx : mn;
  }

  *(float2*)(out + (size_t)ray * NS + 2*lane) = make_float2(s0, s1);
}

extern "C" void kernel_launch(void* const* d_in, const int* in_sizes, int n_in,
                              void* d_out, int out_size, void* d_ws, size_t ws_size,
                              hipStream_t stream) {
  const float* extrinsic = (const float*)d_in[0];
  const float* intrinsic = (const float*)d_in[1];
  const float* Rm        = (const float*)d_in[2];
  const float* Tm        = (const float*)d_in[3];
  const float* Sm        = (const float*)d_in[4];
  const float* density   = (const float*)d_in[5];
  const float* uin       = (const float*)d_in[6];
  float* out = (float*)d_out;

  const int nrays  = in_sizes[5] / NS;               // 262144
  const int blocks = (nrays + WPB - 1) / WPB;        // one ray per wave32
  nerf_importance_kernel<<<blocks, 256, 0, stream>>>(
      extrinsic, intrinsic, Rm, Tm, Sm, density, uin, out, nrays);
}